// GCN8_6279242187095
// MI455X (gfx1250) — compile-verified
//
#include <hip/hip_runtime.h>

#define NN 10000
#define NE 320000
#define MAXD 256

typedef float v2f __attribute__((ext_vector_type(2)));
typedef float v8f __attribute__((ext_vector_type(8)));

// ---------------- degree / norm precompute ----------------

__global__ void zero_f32(float* __restrict__ p, int n) {
  int i = blockIdx.x * blockDim.x + threadIdx.x;
  if (i < n) p[i] = 0.0f;
}

__global__ void deg_count(const int* __restrict__ dst, float* __restrict__ deg, int e) {
  int i = blockIdx.x * blockDim.x + threadIdx.x;
  if (i < e) atomicAdd(&deg[dst[i]], 1.0f);
}

__global__ void dinv_kernel(const float* __restrict__ deg, float* __restrict__ dinv, int n) {
  int i = blockIdx.x * blockDim.x + threadIdx.x;
  if (i < n) dinv[i] = rsqrtf(deg[i] + 1.0f);  // +1 for self loop; deg>=1 always
}

__global__ void norm_kernel(const int* __restrict__ src, const int* __restrict__ dst,
                            const float* __restrict__ dinv, float* __restrict__ norm, int e) {
  int i = blockIdx.x * blockDim.x + threadIdx.x;
  if (i < e) norm[i] = dinv[src[i]] * dinv[dst[i]];
}

// ---------------- WMMA fp32 GEMM: C[MxNc] = A[MxK] @ W[KxNc] ----------------
// One wave (32 threads) per 16x16 output tile. K multiple of 4, Nc multiple of 16,
// gridDim.x*16 == M exactly (no bounds checks -> EXEC all ones for WMMA).

__global__ void gemm16_wmma(const float* __restrict__ A, const float* __restrict__ W,
                            float* __restrict__ C, int K, int Nc) {
  const int lane = threadIdx.x;        // 0..31
  const int half = lane >> 4;          // 0: K pair {k0,k0+1}, 1: {k0+2,k0+3}
  const int l    = lane & 15;
  const long row = (long)blockIdx.x * 16 + l;   // A row held by this lane
  const int  col = blockIdx.y * 16 + l;         // W column held by this lane
  const float* __restrict__ arow = A + row * K;

  v8f acc = {};
  for (int k0 = 0; k0 < K; k0 += 4) {
    const int ka = k0 + half * 2;
    v2f a, b;
    a.x = arow[ka];
    a.y = arow[ka + 1];
    b.x = W[(long)ka * Nc + col];
    b.y = W[(long)(ka + 1) * Nc + col];
    acc = __builtin_amdgcn_wmma_f32_16x16x4_f32(
        /*neg_a=*/false, a, /*neg_b=*/false, b,
        /*c_mod=*/(short)0, acc, /*reuse_a=*/false, /*reuse_b=*/false);
  }

  // C/D layout: VGPR r -> M = tile*16 + r (lanes 0-15) or + r + 8 (lanes 16-31)
  const long orow = (long)blockIdx.x * 16 + half * 8;
#pragma unroll
  for (int r = 0; r < 8; ++r)
    C[(orow + r) * Nc + col] = acc[r];
}

// ---------------- scatter pipeline ----------------

// agg = m * dinv^2 (self-loop contribution), float4-vectorized.
// cshift = log2(D/4) so node = i >> cshift.
__global__ void self_loop_init(const float4* __restrict__ m4, const float* __restrict__ dinv,
                               float4* __restrict__ agg4, int n4, int cshift) {
  int i = blockIdx.x * blockDim.x + threadIdx.x;
  if (i >= n4) return;
  float s = dinv[i >> cshift];
  s = s * s;
  float4 v = m4[i];
  v.x *= s; v.y *= s; v.z *= s; v.w *= s;
  agg4[i] = v;
}

// agg[dst] += m[src] * norm[e], one float4 chunk per thread, f32 global atomics (L2).
__global__ void edge_scatter(const float4* __restrict__ m4, const float* __restrict__ norm,
                             const int* __restrict__ src, const int* __restrict__ dst,
                             float* __restrict__ agg, int total, int cshift, int d) {
  int t = blockIdx.x * blockDim.x + threadIdx.x;
  if (t >= total) return;
  const int e = t >> cshift;
  const int c = t & ((1 << cshift) - 1);
  const int s = src[e];
  const int q = dst[e];
  const float w = norm[e];
  float4 v = m4[s * (d >> 2) + c];
  float* o = agg + (long)q * d + (c << 2);
  atomicAdd(o + 0, v.x * w);
  atomicAdd(o + 1, v.y * w);
  atomicAdd(o + 2, v.z * w);
  atomicAdd(o + 3, v.w * w);
}

// h = relu(agg + b); d is a power of two -> dmask = d-1.
__global__ void bias_relu(const float* __restrict__ agg, const float* __restrict__ b,
                          float* __restrict__ h, int total, int dmask) {
  int i = blockIdx.x * blockDim.x + threadIdx.x;
  if (i >= total) return;
  float v = agg[i] + b[i & dmask];
  h[i] = v > 0.0f ? v : 0.0f;
}

// out = h @ Wr + br  (d=32, OUT=1)
__global__ void readout(const float* __restrict__ h, const float* __restrict__ Wr,
                        const float* __restrict__ br, float* __restrict__ out, int n, int d) {
  int i = blockIdx.x * blockDim.x + threadIdx.x;
  if (i >= n) return;
  float acc = br[0];
  const float* row = h + (long)i * d;
#pragma unroll 8
  for (int k = 0; k < d; ++k) acc += row[k] * Wr[k];
  out[i] = acc;
}

// ---------------- launcher ----------------

extern "C" void kernel_launch(void* const* d_in, const int* in_sizes, int n_in,
                              void* d_out, int out_size, void* d_ws, size_t ws_size,
                              hipStream_t stream) {
  (void)in_sizes; (void)n_in; (void)out_size; (void)ws_size;
  const int N = NN, E = NE;
  static const int dims[9] = {128, 256, 256, 256, 128, 128, 64, 64, 32};

  const float* x  = (const float*)d_in[0];
  const int* ei   = (const int*)d_in[1];
  const int* src  = ei;          // edge_index[0]
  const int* dst  = ei + E;      // edge_index[1]

  // workspace layout (floats): dinv[N] | norm[E] | m[N*256] | agg[N*256] | h[N*256]
  float* ws   = (float*)d_ws;
  float* dinv = ws;                      ws += N;
  float* norm = ws;                      ws += E;
  float* mbuf = ws;                      ws += (size_t)N * MAXD;
  float* agg  = ws;                      ws += (size_t)N * MAXD;
  float* hbuf = ws;

  const int TB = 256;

  // degree -> dinv -> per-edge norm (norm reused by all 8 layers)
  zero_f32  <<<(N + TB - 1) / TB, TB, 0, stream>>>(agg, N);          // agg[0..N) as deg scratch
  deg_count <<<(E + TB - 1) / TB, TB, 0, stream>>>(dst, agg, E);
  dinv_kernel<<<(N + TB - 1) / TB, TB, 0, stream>>>(agg, dinv, N);
  norm_kernel<<<(E + TB - 1) / TB, TB, 0, stream>>>(src, dst, dinv, norm, E);

  const float* hin = x;
  for (int i = 0; i < 8; ++i) {
    const int K = dims[i];
    const int D = dims[i + 1];
    const float* W = (const float*)d_in[2 + 2 * i];
    const float* b = (const float*)d_in[3 + 2 * i];

    dim3 grid(N / 16, D / 16);           // 10000 % 16 == 0, D % 16 == 0
    gemm16_wmma<<<grid, 32, 0, stream>>>(hin, W, mbuf, K, D);

    const int cshift = __builtin_ctz(D / 4);
    const int n4 = N * (D / 4);
    self_loop_init<<<(n4 + TB - 1) / TB, TB, 0, stream>>>(
        (const float4*)mbuf, dinv, (float4*)agg, n4, cshift);

    const int total = E * (D / 4);
    edge_scatter<<<(total + TB - 1) / TB, TB, 0, stream>>>(
        (const float4*)mbuf, norm, src, dst, agg, total, cshift, D);

    const int nt = N * D;
    bias_relu<<<(nt + TB - 1) / TB, TB, 0, stream>>>(agg, b, hbuf, nt, D - 1);

    hin = hbuf;
  }

  const float* Wr = (const float*)d_in[18];
  const float* br = (const float*)d_in[19];
  readout<<<(N + TB - 1) / TB, TB, 0, stream>>>(hin, Wr, br, (float*)d_out, N, dims[8]);
}